// ConcatAttn_83296595739240
// MI455X (gfx1250) — compile-verified
//
#include <hip/hip_runtime.h>
#include <hip/hip_bf16.h>
#include <math.h>

// Problem constants (from reference): H=1024, B=16, T=2048
#define HD 1024
#define BD 16
#define TD 2048

typedef __attribute__((ext_vector_type(2))) float v2f;
typedef __attribute__((ext_vector_type(8))) float v8f;

// ---------------------------------------------------------------------------
// Kernel 1a: weff[k] = sum_h v[h] * W[h, k]   for k in [0, 2H)
// W is [H, 2H] row-major. Threads = k -> fully coalesced column sweep.
// Reads 8 MB once (~0.35 us at 23.3 TB/s).
// ---------------------------------------------------------------------------
__global__ void __launch_bounds__(256)
fold_vW_kernel(const float* __restrict__ W, const float* __restrict__ v,
               float* __restrict__ weff) {
    const int k = blockIdx.x * 256 + threadIdx.x;   // 0..2047
    float acc = 0.f;
#pragma unroll 4
    for (int h = 0; h < HD; ++h)
        acc = fmaf(v[h], W[(size_t)h * (2 * HD) + k], acc);
    weff[k] = acc;
}

// ---------------------------------------------------------------------------
// Kernel 1b: cb[b] = sum_h hidden[b,h]*weff[h] + sum_h v[h]*bias[h]
// (softmax-invariant per-b constant, kept for faithfulness to the reference)
// ---------------------------------------------------------------------------
__global__ void __launch_bounds__(256)
fold_hidden_kernel(const float* __restrict__ hidden, const float* __restrict__ weff,
                   const float* __restrict__ v, const float* __restrict__ bias,
                   float* __restrict__ cb) {
    __shared__ float red[256];
    const int b = blockIdx.x;
    float acc = 0.f;
    for (int h = threadIdx.x; h < HD; h += 256)
        acc = fmaf(hidden[b * HD + h], weff[h], fmaf(v[h], bias[h], acc));
    red[threadIdx.x] = acc;
    __syncthreads();
    for (int s = 128; s > 0; s >>= 1) {
        if (threadIdx.x < s) red[threadIdx.x] += red[threadIdx.x + s];
        __syncthreads();
    }
    if (threadIdx.x == 0) cb[b] = red[0];
}

// ---------------------------------------------------------------------------
// Kernel 2: streaming dot via V_WMMA_F32_16X16X4_F32.
// One wave owns one t-tile = 16 contiguous rows of enc ([T,B,H]: t fixed,
// b=0..15 -> one contiguous 64KB block). Accumulate over K in steps of 4.
//
// fp32 fragment layouts (ISA 7.12.2):
//   A (16x4, 2 VGPRs): lane m (0-15) / m+16 hold K = 2*hi, 2*hi+1
//   B (4x16): symmetric; broadcast w2 chunk across all 16 columns
//   C/D (16x16, 8 VGPRs): VGPR r, half hi  ->  row M = r + 8*hi (all N equal)
// ---------------------------------------------------------------------------
__global__ void __launch_bounds__(256)
dot_wmma_kernel(const float* __restrict__ enc, const float* __restrict__ weff,
                const float* __restrict__ cb, float* __restrict__ s) {
    const int lane = threadIdx.x & 31;
    const int wave = threadIdx.x >> 5;
    const int t    = blockIdx.x * 8 + wave;      // 0..T-1
    const int m    = lane & 15;                  // row within tile (= b)
    const int hi   = lane >> 4;                  // half-wave select (K phase)

    const float* __restrict__ row = enc + (size_t)t * (BD * HD)
                                        + (size_t)m * HD + 2 * hi;
    const float* __restrict__ w2  = weff + HD + 2 * hi;   // second half of Wᵀv

    v8f acc = {};
#pragma unroll 8
    for (int k = 0; k < HD; k += 4) {
        const float2 av = *(const float2*)(row + k);      // 8B-aligned
        const float2 wv = *(const float2*)(w2 + k);       // 8B-aligned (L2-hot)
        v2f a;  a.x = av.x;  a.y = av.y;
        v2f bb; bb.x = wv.x; bb.y = wv.y;
        acc = __builtin_amdgcn_wmma_f32_16x16x4_f32(
                  /*neg_a=*/false, a, /*neg_b=*/false, bb,
                  /*c_mod=*/(short)0, acc, /*reuse_a=*/false, /*reuse_b=*/false);
    }

    // Every column of D is identical; lanes 0 and 16 each own 8 rows.
    if (m == 0) {
#pragma unroll
        for (int r = 0; r < 8; ++r) {
            const int b = r + 8 * hi;
            s[(size_t)b * TD + t] = acc[r] + cb[b];
        }
    }
}

// ---------------------------------------------------------------------------
// Kernel 3: softmax over t for each b (16 rows x 2048). One block per row.
// ---------------------------------------------------------------------------
__global__ void __launch_bounds__(256)
softmax_kernel(const float* __restrict__ s, float* __restrict__ out) {
    __shared__ float red[256];
    const int b = blockIdx.x;
    const float* __restrict__ srow = s + (size_t)b * TD;
    float* __restrict__ orow = out + (size_t)b * TD;

    float vals[TD / 256];
    float lmax = -INFINITY;
#pragma unroll
    for (int i = 0; i < TD / 256; ++i) {
        vals[i] = srow[threadIdx.x + i * 256];
        lmax = fmaxf(lmax, vals[i]);
    }
    red[threadIdx.x] = lmax;
    __syncthreads();
    for (int st = 128; st > 0; st >>= 1) {
        if (threadIdx.x < st)
            red[threadIdx.x] = fmaxf(red[threadIdx.x], red[threadIdx.x + st]);
        __syncthreads();
    }
    const float gmax = red[0];
    __syncthreads();

    float lsum = 0.f;
#pragma unroll
    for (int i = 0; i < TD / 256; ++i) {
        vals[i] = __expf(vals[i] - gmax);
        lsum += vals[i];
    }
    red[threadIdx.x] = lsum;
    __syncthreads();
    for (int st = 128; st > 0; st >>= 1) {
        if (threadIdx.x < st) red[threadIdx.x] += red[threadIdx.x + st];
        __syncthreads();
    }
    const float inv = 1.0f / red[0];
#pragma unroll
    for (int i = 0; i < TD / 256; ++i)
        orow[threadIdx.x + i * 256] = vals[i] * inv;
}

// ---------------------------------------------------------------------------
// Inputs (setup_inputs order): hidden [1,B,H], encoder_output [T,B,H],
// attn_W [H,2H], attn_b [H], v [H].  Output: [B,1,T] fp32.
// Workspace layout (fp32): weff[2048] | cb[16] pad | scores[B*T] @ +4096.
// Total ws use = (4096 + 32768) * 4 = 147456 bytes.
// ---------------------------------------------------------------------------
extern "C" void kernel_launch(void* const* d_in, const int* in_sizes, int n_in,
                              void* d_out, int out_size, void* d_ws, size_t ws_size,
                              hipStream_t stream) {
    const float* hidden = (const float*)d_in[0];
    const float* enc    = (const float*)d_in[1];
    const float* W      = (const float*)d_in[2];
    const float* bias   = (const float*)d_in[3];
    const float* v      = (const float*)d_in[4];
    float* out  = (float*)d_out;

    float* weff = (float*)d_ws;          // 2048 floats
    float* cb   = weff + 2048;           // 16 floats
    float* sbuf = weff + 4096;           // 32768 floats

    fold_vW_kernel   <<<(2 * HD) / 256, 256, 0, stream>>>(W, v, weff);
    fold_hidden_kernel<<<BD, 256, 0, stream>>>(hidden, weff, v, bias, cb);
    dot_wmma_kernel  <<<TD / 8, 256, 0, stream>>>(enc, weff, cb, sbuf);
    softmax_kernel   <<<BD, 256, 0, stream>>>(sbuf, out);
}